// simple_RGGC_87789131531002
// MI455X (gfx1250) — compile-verified
//
#include <hip/hip_runtime.h>
#include <hip/hip_bf16.h>

typedef float v2f __attribute__((ext_vector_type(2)));
typedef float v8f __attribute__((ext_vector_type(8)));
typedef unsigned int u32x4 __attribute__((ext_vector_type(4)));
typedef int i32x4 __attribute__((ext_vector_type(4)));
typedef int i32x8 __attribute__((ext_vector_type(8)));

#define NN 50000
#define NE 800000
#define DD 64
#define NL 5
#define NG 256
#define NC 10
#define BN_EPS 1e-5f
#define ROWS_PER_STATS_BLOCK 200

// ---------------------------------------------------------------------------
// TDM: DMA a row-major [64 x 64] f32 matrix from global into LDS.
// D# layout per CDNA5 ISA ch.8 (CDNA5 ISA):
//   group0: [1:0]=count=1, [63:32]=lds_addr, [120:64]=global_addr, [127:126]=type=2
//   group1: [17:16]=data_size(2 -> 4B), [79:48]=tensor_dim0=64,
//           [111:80]=tensor_dim1=64, [127:112]=tile_dim0=64, [143:128]=tile_dim1=64,
//           [207:160]=tensor_dim0_stride=64; everything else 0 (no multicast/pad/iterate)
//   groups 2/3: zero (tensor is 2-D)
// ---------------------------------------------------------------------------
__device__ __forceinline__ void tdm_load_w64x64(const float* gW, unsigned lds_byte_addr) {
    const unsigned long long ga = (unsigned long long)(const void*)gW;

    u32x4 g0;
    g0.x = 1u;                                        // count=1, is_restore=0, gather=0
    g0.y = lds_byte_addr;                             // lds_addr
    g0.z = (unsigned)(ga & 0xFFFFFFFFull);            // global_addr[31:0]
    g0.w = (unsigned)((ga >> 32) & 0x01FFFFFFull)     // global_addr[56:32]
           | 0x80000000u;                             // type=2 ("image")

    i32x8 g1;
    g1.s0 = (int)(2u << 16);                          // data_size = 4 bytes
    g1.s1 = (int)(64u << 16);                         // tensor_dim0[15:0] at bits 63:48
    g1.s2 = (int)(64u << 16);                         // tensor_dim1[15:0] at bits 111:96 half (bits 95:80)
    g1.s3 = (int)(64u << 16);                         // tile_dim0=64 at bits 127:112
    g1.s4 = (int)64;                                  // tile_dim1=64 at bits 143:128; tile_dim2=0
    g1.s5 = (int)64;                                  // tensor_dim0_stride[31:0]=64 (bits 191:160)
    g1.s6 = 0;                                        // stride0 hi | tensor_dim1_stride lo
    g1.s7 = 0;

    i32x4 gz4 = {0, 0, 0, 0};
    i32x8 gz8 = {0, 0, 0, 0, 0, 0, 0, 0};

    __builtin_amdgcn_tensor_load_to_lds(g0, g1, gz4, gz4, gz8, 0);
    __builtin_amdgcn_s_wait_tensorcnt(0);
}

// ---------------------------------------------------------------------------
// Fused 4-way GEMM: out = H @ W + b for (K,Q,V,skip->AGG), selected by
// blockIdx.y. W is staged to LDS via the Tensor Data Mover; each wave then
// computes a 16x64 output strip via 64 V_WMMA_F32_16X16X4_F32 ops.
// ---------------------------------------------------------------------------
__global__ void rggc_gemm4_wmma(const float* __restrict__ H,
                                const float* __restrict__ Wk, const float* __restrict__ Wq,
                                const float* __restrict__ Wv, const float* __restrict__ Ws,
                                const float* __restrict__ bk, const float* __restrict__ bq,
                                const float* __restrict__ bv, const float* __restrict__ bs,
                                float* __restrict__ Ko, float* __restrict__ Qo,
                                float* __restrict__ Vo, float* __restrict__ Ao,
                                int nrows) {
    __shared__ float sW[DD * DD];  // 16 KB

    const int lane = threadIdx.x & 31;
    const int wave = threadIdx.x >> 5;

    const float* W;
    const float* bias;
    float* O;
    switch (blockIdx.y) {
        case 0:  W = Wk; bias = bk; O = Ko; break;
        case 1:  W = Wq; bias = bq; O = Qo; break;
        case 2:  W = Wv; bias = bv; O = Vo; break;
        default: W = Ws; bias = bs; O = Ao; break;
    }

    // Wave 0 issues one TDM DMA for the whole 16 KB weight tile. TDM ignores
    // EXEC and issues once per wave, so restrict to a single wave.
    if (wave == 0) {
        tdm_load_w64x64(W, (unsigned)(unsigned long long)(void*)sW);
    }
    __syncthreads();

    const int row0 = (blockIdx.x * 8 + wave) * 16;
    if (row0 >= nrows) return;

    const int rm = lane & 15;          // row within tile (A), col within tile (B/C)
    const int kh = (lane >> 4) * 2;    // lanes 0-15 -> K pair {0,1}; 16-31 -> {2,3}

    v8f acc0 = {}, acc1 = {}, acc2 = {}, acc3 = {};

    const float* hrow = H + (size_t)(row0 + rm) * DD;

#pragma unroll
    for (int kk = 0; kk < 16; ++kk) {
        const int kb = kk * 4;
        v2f a;
        a.x = hrow[kb + kh];
        a.y = hrow[kb + kh + 1];

        const float* w0 = sW + (kb + kh) * DD + rm;
        const float* w1 = sW + (kb + kh + 1) * DD + rm;

        v2f b0; b0.x = w0[0];  b0.y = w1[0];
        v2f b1; b1.x = w0[16]; b1.y = w1[16];
        v2f b2; b2.x = w0[32]; b2.y = w1[32];
        v2f b3; b3.x = w0[48]; b3.y = w1[48];

        acc0 = __builtin_amdgcn_wmma_f32_16x16x4_f32(false, a, false, b0, (short)0, acc0, false, false);
        acc1 = __builtin_amdgcn_wmma_f32_16x16x4_f32(false, a, false, b1, (short)0, acc1, false, false);
        acc2 = __builtin_amdgcn_wmma_f32_16x16x4_f32(false, a, false, b2, (short)0, acc2, false, false);
        acc3 = __builtin_amdgcn_wmma_f32_16x16x4_f32(false, a, false, b3, (short)0, acc3, false, false);
    }

    // C/D layout: vgpr r -> M = r (lanes 0-15) or r+8 (lanes 16-31); N = rm.
    const int mbase = (lane >> 4) * 8;
    const float bi0 = bias[rm];
    const float bi1 = bias[16 + rm];
    const float bi2 = bias[32 + rm];
    const float bi3 = bias[48 + rm];
#pragma unroll
    for (int r = 0; r < 8; ++r) {
        float* orow = O + (size_t)(row0 + mbase + r) * DD + rm;
        orow[0]  = acc0[r] + bi0;
        orow[16] = acc1[r] + bi1;
        orow[32] = acc2[r] + bi2;
        orow[48] = acc3[r] + bi3;
    }
}

// ---------------------------------------------------------------------------
// Edge kernel: one wave per edge, lane = feature pair.
// AGG[dst] += sigmoid(K[dst] + Q[src]) * V[src]
// ---------------------------------------------------------------------------
__global__ void rggc_edge(const int* __restrict__ ei,
                          const float* __restrict__ K, const float* __restrict__ Q,
                          const float* __restrict__ V, float* __restrict__ AGG,
                          int nedges) {
    const int lane = threadIdx.x & 31;
    const int wave = threadIdx.x >> 5;
    const int e = blockIdx.x * 8 + wave;
    if (e >= nedges) return;

    const int src = ei[e];
    const int dst = ei[nedges + e];
    const int f = lane * 2;

    const float2 kd = *(const float2*)(K + (size_t)dst * DD + f);
    const float2 qs = *(const float2*)(Q + (size_t)src * DD + f);
    const float2 vs = *(const float2*)(V + (size_t)src * DD + f);

    const float ex = 1.0f / (1.0f + __expf(-(kd.x + qs.x)));
    const float ey = 1.0f / (1.0f + __expf(-(kd.y + qs.y)));

    float* dstp = AGG + (size_t)dst * DD + f;
    atomicAdd(dstp, ex * vs.x);
    atomicAdd(dstp + 1, ey * vs.y);
}

// ---------------------------------------------------------------------------
// Zero a float buffer.
// ---------------------------------------------------------------------------
__global__ void rggc_zero(float* __restrict__ p, int n) {
    int i = blockIdx.x * blockDim.x + threadIdx.x;
    int stride = gridDim.x * blockDim.x;
    for (; i < n; i += stride) p[i] = 0.0f;
}

// ---------------------------------------------------------------------------
// In-place ReLU + per-column sum / sum-of-squares (LDS partials, then one
// atomic per column per block). stats[0:64]=sum, stats[64:128]=sumsq.
// ---------------------------------------------------------------------------
__global__ void rggc_relu_stats(float* __restrict__ A, float* __restrict__ stats, int nrows) {
    __shared__ float s_sum[256];
    __shared__ float s_sq[256];
    const int t = threadIdx.x;
    const int col = t & 63;
    const int rsub = t >> 6;  // 0..3

    const int rb = blockIdx.x * ROWS_PER_STATS_BLOCK;
    int re = rb + ROWS_PER_STATS_BLOCK;
    if (re > nrows) re = nrows;

    float sum = 0.0f, sq = 0.0f;
    for (int r = rb + rsub; r < re; r += 4) {
        float* p = A + (size_t)r * DD + col;
        float x = *p;
        x = fmaxf(x, 0.0f);
        *p = x;
        sum += x;
        sq += x * x;
    }
    s_sum[t] = sum;
    s_sq[t] = sq;
    __syncthreads();
    if (t < 64) {
        float S = s_sum[t] + s_sum[t + 64] + s_sum[t + 128] + s_sum[t + 192];
        float Q2 = s_sq[t] + s_sq[t + 64] + s_sq[t + 128] + s_sq[t + 192];
        atomicAdd(&stats[t], S);
        atomicAdd(&stats[64 + t], Q2);
    }
}

// ---------------------------------------------------------------------------
// BatchNorm normalize: Hout = (A - mean) * rsqrt(var + eps) * gamma + beta
// ---------------------------------------------------------------------------
__global__ void rggc_bn(const float* __restrict__ A, const float* __restrict__ stats,
                        const float* __restrict__ gamma, const float* __restrict__ beta,
                        float* __restrict__ Hout, int nrows) {
    const float inv_n = 1.0f / (float)nrows;
    int i = blockIdx.x * blockDim.x + threadIdx.x;
    const int stride = gridDim.x * blockDim.x;
    const int total = nrows * DD;
    for (; i < total; i += stride) {
        const int col = i & 63;
        const float mean = stats[col] * inv_n;
        const float var = stats[64 + col] * inv_n - mean * mean;
        const float x = A[i];
        Hout[i] = (x - mean) * rsqrtf(var + BN_EPS) * gamma[col] + beta[col];
    }
}

// ---------------------------------------------------------------------------
// Global mean pool (sum + count phase): one wave per node.
// ---------------------------------------------------------------------------
__global__ void rggc_pool(const float* __restrict__ H, const int* __restrict__ batch,
                          float* __restrict__ pooled, float* __restrict__ counts,
                          int nrows) {
    const int lane = threadIdx.x & 31;
    const int wave = threadIdx.x >> 5;
    const int node = blockIdx.x * 8 + wave;
    if (node >= nrows) return;
    const int g = batch[node];
    const int f = lane * 2;
    const float2 h = *(const float2*)(H + (size_t)node * DD + f);
    float* pp = pooled + (size_t)g * DD + f;
    atomicAdd(pp, h.x);
    atomicAdd(pp + 1, h.y);
    if (lane == 0) atomicAdd(&counts[g], 1.0f);
}

// ---------------------------------------------------------------------------
// Final: pooled mean @ Wlin + blin, softmax. One thread per graph.
// ---------------------------------------------------------------------------
__global__ void rggc_head(const float* __restrict__ pooled, const float* __restrict__ counts,
                          const float* __restrict__ Wlin, const float* __restrict__ blin,
                          float* __restrict__ out) {
    const int g = blockIdx.x * blockDim.x + threadIdx.x;
    if (g >= NG) return;
    const float inv_c = 1.0f / fmaxf(counts[g], 1.0f);

    float logits[NC];
#pragma unroll
    for (int c = 0; c < NC; ++c) logits[c] = blin[c];
    const float* pg = pooled + (size_t)g * DD;
    for (int k = 0; k < DD; ++k) {
        const float x = pg[k] * inv_c;
        const float* wr = Wlin + (size_t)k * NC;
#pragma unroll
        for (int c = 0; c < NC; ++c) logits[c] += x * wr[c];
    }
    float m = logits[0];
#pragma unroll
    for (int c = 1; c < NC; ++c) m = fmaxf(m, logits[c]);
    float s = 0.0f;
#pragma unroll
    for (int c = 0; c < NC; ++c) {
        logits[c] = __expf(logits[c] - m);
        s += logits[c];
    }
    const float inv_s = 1.0f / s;
#pragma unroll
    for (int c = 0; c < NC; ++c) out[g * NC + c] = logits[c] * inv_s;
}

// ---------------------------------------------------------------------------
extern "C" void kernel_launch(void* const* d_in, const int* in_sizes, int n_in,
                              void* d_out, int out_size, void* d_ws, size_t ws_size,
                              hipStream_t stream) {
    const float* X       = (const float*)d_in[0];
    const int*   ei      = (const int*)d_in[1];   // [2, NE]
    const int*   batch   = (const int*)d_in[2];   // [NN]
    const float* Wk      = (const float*)d_in[3]; // [NL, 64, 64]
    const float* Wq      = (const float*)d_in[4];
    const float* Wv      = (const float*)d_in[5];
    const float* Ws      = (const float*)d_in[6];
    const float* bk      = (const float*)d_in[7]; // [NL, 64]
    const float* bq      = (const float*)d_in[8];
    const float* bv      = (const float*)d_in[9];
    const float* bconv   = (const float*)d_in[10];
    const float* gamma   = (const float*)d_in[11];
    const float* beta    = (const float*)d_in[12];
    const float* Wlin    = (const float*)d_in[13]; // [64, 10]
    const float* blin    = (const float*)d_in[14]; // [10]
    float* out = (float*)d_out;

    // Workspace layout (floats)
    float* H      = (float*)d_ws;                 // NN*DD
    float* Kb     = H + (size_t)NN * DD;          // NN*DD
    float* Qb     = Kb + (size_t)NN * DD;         // NN*DD
    float* Vb     = Qb + (size_t)NN * DD;         // NN*DD
    float* AGG    = Vb + (size_t)NN * DD;         // NN*DD
    float* stats  = AGG + (size_t)NN * DD;        // 128
    float* pooled = stats + 128;                  // NG*DD
    float* counts = pooled + (size_t)NG * DD;     // NG

    // H = X
    (void)hipMemcpyAsync(H, X, (size_t)NN * DD * sizeof(float), hipMemcpyDeviceToDevice, stream);

    const int gemm_blocks = ((NN / 16) + 7) / 8;        // waves of 16 rows, 8 per block
    const int edge_blocks = (NE + 7) / 8;
    const int stats_blocks = (NN + ROWS_PER_STATS_BLOCK - 1) / ROWS_PER_STATS_BLOCK;

    for (int l = 0; l < NL; ++l) {
        const size_t wo = (size_t)l * DD * DD;
        const size_t bo = (size_t)l * DD;

        // K,Q,V and skip-path (H@Ws + bconv -> AGG)
        rggc_gemm4_wmma<<<dim3(gemm_blocks, 4), 256, 0, stream>>>(
            H, Wk + wo, Wq + wo, Wv + wo, Ws + wo,
            bk + bo, bq + bo, bv + bo, bconv + bo,
            Kb, Qb, Vb, AGG, NN);

        // gated aggregation into AGG
        rggc_edge<<<edge_blocks, 256, 0, stream>>>(ei, Kb, Qb, Vb, AGG, NE);

        // relu + BN
        rggc_zero<<<1, 128, 0, stream>>>(stats, 128);
        rggc_relu_stats<<<stats_blocks, 256, 0, stream>>>(AGG, stats, NN);
        rggc_bn<<<2048, 256, 0, stream>>>(AGG, stats, gamma + bo, beta + bo, H, NN);
    }

    // mean pool + head
    rggc_zero<<<80, 256, 0, stream>>>(pooled, NG * DD + NG);  // pooled and counts contiguous
    rggc_pool<<<(NN + 7) / 8, 256, 0, stream>>>(H, batch, pooled, counts, NN);
    rggc_head<<<1, NG, 0, stream>>>(pooled, counts, Wlin, blin, out);
}